// GAT_24000277250315
// MI455X (gfx1250) — compile-verified
//
#include <hip/hip_runtime.h>

typedef float v2f __attribute__((ext_vector_type(2)));
typedef float v8f __attribute__((ext_vector_type(8)));
typedef int   ivec4 __attribute__((vector_size(16)));
typedef __attribute__((address_space(1))) ivec4* gvec4p;   // global int4*
typedef __attribute__((address_space(3))) ivec4* lvec4p;   // LDS int4*

#define NNEI 50
#define NS   51          // neighbors + self
#define DD   100
#define RS   104         // LDS row stride in kernel 1 (float4 aligned)
#define MT   16          // M tile
#define NT   7           // ceil(100/16) N tiles
#define NPAD (NT * MT)   // 112 padded N

#if defined(__gfx1250__) && __has_builtin(__builtin_amdgcn_global_load_async_to_lds_b128) && __has_builtin(__builtin_amdgcn_s_wait_asynccnt)
#define USE_ASYNC_LDS 1
#else
#define USE_ASYNC_LDS 0
#endif

// ---------------- Kernel 1: attention context --------------------------------
// One block per row b. Stage the 51x100 tile (50 neighbors + query) in LDS
// (async global->LDS when available), compute scores, softmax, weighted sum.
// Writes ctx row into ctx_out (aliases d_out; kernel 2 rewrites it in place).
__global__ __launch_bounds__(128) void gat_context_kernel(
    const float* __restrict__ self_vecs,    // B x D
    const float* __restrict__ neigh_vecs,   // B x N x D
    float* __restrict__ ctx_out)            // B x D
{
    __shared__ float tile[NS * RS];         // ~21.2 KB
    __shared__ float sc[NS];
    __shared__ float sinv;

    const int b   = blockIdx.x;
    const int tid = threadIdx.x;

    const float4* nsrc = (const float4*)(neigh_vecs + (size_t)b * (NNEI * DD));
    const float4* ssrc = (const float4*)(self_vecs + (size_t)b * DD);

#if USE_ASYNC_LDS
    // CDNA5 async copy: global -> LDS without VGPR round trip (ASYNCcnt).
    for (int i = tid; i < NNEI * (DD / 4); i += 128) {
        int n  = i / (DD / 4);
        int d4 = i % (DD / 4);
        __builtin_amdgcn_global_load_async_to_lds_b128(
            (gvec4p)(nsrc + i),
            (lvec4p)&tile[n * RS + d4 * 4],
            0, 0);
    }
    for (int i = tid; i < DD / 4; i += 128) {
        __builtin_amdgcn_global_load_async_to_lds_b128(
            (gvec4p)(ssrc + i),
            (lvec4p)&tile[NNEI * RS + i * 4],
            0, 0);
    }
    __builtin_amdgcn_s_wait_asynccnt(0);
#else
    for (int i = tid; i < NNEI * (DD / 4); i += 128) {
        int n  = i / (DD / 4);
        int d4 = i % (DD / 4);
        *(float4*)&tile[n * RS + d4 * 4] = nsrc[i];
    }
    for (int i = tid; i < DD / 4; i += 128) {
        *(float4*)&tile[NNEI * RS + i * 4] = ssrc[i];
    }
#endif
    __syncthreads();

    // Scores: s[n] = dot(row n, query row 50), n = 0..50 (incl. q.q)
    if (tid < NS) {
        const float* q = &tile[NNEI * RS];
        const float* r = &tile[tid * RS];
        float s = 0.f;
        #pragma unroll 4
        for (int d = 0; d < DD; ++d) s = fmaf(r[d], q[d], s);
        sc[tid] = s;
    }
    __syncthreads();

    // Softmax over 51 values: trivial serial work vs 20 KB of HBM per block
    if (tid == 0) {
        float m = sc[0];
        for (int n = 1; n < NS; ++n) m = fmaxf(m, sc[n]);
        float sum = 0.f;
        for (int n = 0; n < NS; ++n) { float e = __expf(sc[n] - m); sc[n] = e; sum += e; }
        sinv = 1.f / sum;
    }
    __syncthreads();

    // Context: ctx[d] = (1/sum) * sum_n e[n] * tile[n][d]
    if (tid < DD) {
        float acc = 0.f;
        #pragma unroll 4
        for (int n = 0; n < NS; ++n) acc = fmaf(sc[n], tile[n * RS + tid], acc);
        ctx_out[(size_t)b * DD + tid] = acc * sinv;
    }
}

// ---------------- Kernel 2: out = relu(ctx @ W) via V_WMMA_F32_16X16X4_F32 ---
// One block per 16-row M tile; 7 waves, one 16-wide N tile each. Both operands
// staged in LDS in K-pair layout so the inner loop is 2x ds_load_b64 + wmma
// with no conditionals (EXEC stays all-1s). W is zero-padded to 112 columns.
__global__ __launch_bounds__(NT * 32) void gat_gemm_relu_kernel(
    const float* __restrict__ weights,  // D x D row-major
    float* __restrict__ io)             // B x D: reads ctx, writes relu(ctx@W)
{
    __shared__ float atp[(DD / 2) * (MT * 2)];     // A pairs: [k/2][m][2], 6.4 KB
    __shared__ float bwp[(DD / 2) * (NPAD * 2)];   // B pairs: [k/2][n][2], 44.8 KB

    const int mbase = blockIdx.x * MT;
    const int tid   = threadIdx.x;

    // Stage 16 ctx rows (contiguous 1600 floats) -> paired A layout
    const float4* src = (const float4*)(io + (size_t)mbase * DD);
    for (int i = tid; i < MT * (DD / 4); i += NT * 32) {
        float4 v = src[i];
        int e  = 4 * i;
        int m  = e / DD;
        int d  = e % DD;        // multiple of 4
        int kk = d >> 1;        // K-pair index of (d, d+1)
        *(v2f*)&atp[kk * (MT * 2) + m * 2]       = (v2f){v.x, v.y};
        *(v2f*)&atp[(kk + 1) * (MT * 2) + m * 2] = (v2f){v.z, v.w};
    }
    // Stage W -> paired, zero-padded layout. Thread t owns column n=t/2, j=t&1.
    {
        const int n = tid >> 1, j = tid & 1;
        for (int kk = 0; kk < DD / 2; ++kk) {
            float w = (n < DD) ? weights[(2 * kk + j) * DD + n] : 0.f;
            bwp[kk * (NPAD * 2) + tid] = w;
        }
    }
    __syncthreads();

    const int wave  = tid >> 5;            // 0..6 -> N tile
    const int lane  = tid & 31;
    const int mrow  = lane & 15;           // A: M index for this lane
    const int sel   = lane >> 4;           // lanes 16-31 hold K+2,K+3
    const int ncol  = wave * MT + (lane & 15);
    const int aoff  = mrow * 2;
    const int boff  = ncol * 2;

    v8f c = {};                            // f32 16x16 accumulator (8 VGPRs)
    #pragma unroll 5
    for (int kp2 = 0; kp2 < DD / 4; ++kp2) {
        const int kp = kp2 * 2 + sel;      // K-pair index for this half-wave
        v2f a  = *(const v2f*)&atp[kp * (MT * 2) + aoff];
        v2f bf = *(const v2f*)&bwp[kp * (NPAD * 2) + boff];
        c = __builtin_amdgcn_wmma_f32_16x16x4_f32(
                false, a, false, bf, (short)0, c, false, false);
    }

    // D layout: VGPR r -> M = r (lanes 0-15) / M = 8+r (lanes 16-31), N = lane&15
    if (ncol < DD) {
        const int mr = mbase + sel * 8;
        #pragma unroll
        for (int r = 0; r < 8; ++r) {
            io[(size_t)(mr + r) * DD + ncol] = fmaxf(c[r], 0.f);
        }
    }
}

// ---------------- Launcher ---------------------------------------------------
extern "C" void kernel_launch(void* const* d_in, const int* in_sizes, int n_in,
                              void* d_out, int out_size, void* d_ws, size_t ws_size,
                              hipStream_t stream) {
    const float* self_vecs  = (const float*)d_in[0];   // B x 100
    const float* neigh_vecs = (const float*)d_in[1];   // B x 50 x 100
    const float* weights    = (const float*)d_in[2];   // 100 x 100
    float* out = (float*)d_out;                        // B x 100

    const int B = in_sizes[0] / DD;                    // 65536

    gat_context_kernel<<<B, 128, 0, stream>>>(self_vecs, neigh_vecs, out);
    gat_gemm_relu_kernel<<<B / MT, NT * 32, 0, stream>>>(weights, out);
}